// FFRFLayer_87076166959109
// MI455X (gfx1250) — compile-verified
//
#include <hip/hip_runtime.h>
#include <hip/hip_bf16.h>

typedef __attribute__((ext_vector_type(2))) float v2f;
typedef __attribute__((ext_vector_type(8))) float v8f;

#define BATCH 64
#define CIN 16
#define H 48
#define W 48
#define KH 5
#define KW 5
#define COUT 128
#define OH 44
#define OW 44
#define P (OH * OW)          // 1936
#define F (CIN * KH * KW)    // 400
#define KCHUNK 40            // F = 10 * 40
#define NCHUNK (F / KCHUNK)  // 10
#define A_STRIDE 42          // padded row stride for As (bank-conflict avoidance)

// ---------------------------------------------------------------- norms ----
__global__ void norm_kernel(const float* __restrict__ x, float* __restrict__ scale) {
    int b = blockIdx.x;
    const float* xb = x + (size_t)b * (CIN * H * W);
    float s = 0.f;
    for (int i = threadIdx.x; i < CIN * H * W; i += 256) {
        float v = xb[i];
        s += v * v;
    }
    __shared__ float sh[256];
    sh[threadIdx.x] = s;
    __syncthreads();
    for (int w = 128; w > 0; w >>= 1) {
        if (threadIdx.x < w) sh[threadIdx.x] += sh[threadIdx.x + w];
        __syncthreads();
    }
    if (threadIdx.x == 0) scale[b] = 1.0f / (sqrtf(sh[0]) + 1e-7f);
}

// ----------------------------------------------------------------- gemm ----
// One workgroup per output position p. Computes (64x400)x(400x128) with
// V_WMMA_F32_16X16X4_F32. Weight chunks are register-double-buffered so the
// HBM stream overlaps the WMMA work. Output written pre-BN to ws in
// (P, B, COUT) order.
__global__ void __launch_bounds__(256, 2)
gemm_kernel(const float* __restrict__ x, const float* __restrict__ kern,
            const float* __restrict__ scale, float* __restrict__ out) {
    __shared__ float As[BATCH * A_STRIDE];  // 64 rows x 40 (+2 pad) K-values
    __shared__ float Bs[KCHUNK * COUT];     // swizzled: (k>>2)*512 + c*4 + (k&3)
    __shared__ float sscale[BATCH];

    const int p = blockIdx.x;
    const int oh = p / OW;
    const int ow = p - oh * OW;
    const int tid = threadIdx.x;
    const int wave = tid >> 5;
    const int lane = tid & 31;
    const int m0 = (wave >> 1) << 4;   // 0,16,32,48
    const int n0 = (wave & 1) << 6;    // 0,64
    const int half = lane >> 4;        // 0 or 1
    const int l15 = lane & 15;

    if (tid < BATCH) sscale[tid] = scale[tid];
    __syncthreads();

    v8f acc[4] = {};
    float4 bv[5];   // staged weight chunk (5 * 256 float4 = 40x128 f32)
    float av[10];   // staged patch chunk (10 * 256 = 64x40 f32)

    const float* Ap = As + (m0 + l15) * A_STRIDE + (half << 1);
    const float* Bp0 = Bs + ((n0 + l15) << 2) + (half << 1);
    const float4* kbase_ptr = (const float4*)(kern + (size_t)p * F * COUT);

    // ---- register-load chunk `kbase` of the weight stream (coalesced b128)
    auto loadB = [&](int kbase) {
#pragma unroll
        for (int u = 0; u < 5; ++u)
            bv[u] = kbase_ptr[(size_t)kbase * (COUT / 4) + tid + u * 256];
    };
    // ---- register-load chunk of the (scaled) patch matrix from x
    auto loadA = [&](int kbase) {
#pragma unroll
        for (int u = 0; u < 10; ++u) {
            int i = tid + u * 256;
            int b = i / KCHUNK;
            int j = i - b * KCHUNK;
            int f = kbase + j;
            int c = f / (KH * KW);
            int r = f - c * (KH * KW);
            int kh = r / KW;
            int kw = r - kh * KW;
            av[u] = x[((size_t)b * CIN + c) * (H * W) + (oh + kh) * W + (ow + kw)] *
                    sscale[b];
        }
    };
    // ---- publish staged registers to LDS (B gets the K-swizzle layout)
    auto storeLDS = [&]() {
#pragma unroll
        for (int u = 0; u < 5; ++u) {
            int i = tid + u * 256;
            int kk = i >> 5;
            int c4 = i & 31;
            int base = ((kk >> 2) << 9) + (c4 << 4) + (kk & 3);
            Bs[base] = bv[u].x;
            Bs[base + 4] = bv[u].y;
            Bs[base + 8] = bv[u].z;
            Bs[base + 12] = bv[u].w;
        }
#pragma unroll
        for (int u = 0; u < 10; ++u) {
            int i = tid + u * 256;
            int b = i / KCHUNK;
            int j = i - b * KCHUNK;
            As[b * A_STRIDE + j] = av[u];
        }
    };

    // prologue: stage chunk 0
    loadB(0);
    loadA(0);
    storeLDS();
    __syncthreads();

    for (int kc = 0; kc < NCHUNK; ++kc) {
        // issue next chunk's global loads; latency hidden under the WMMAs
        if (kc + 1 < NCHUNK) {
            loadB((kc + 1) * KCHUNK);
            loadA((kc + 1) * KCHUNK);
        }
        // compute on current LDS contents
        for (int k = 0; k < KCHUNK; k += 4) {
            v2f a = *(const v2f*)(Ap + k);
            const float* bgrp = Bp0 + ((k >> 2) << 9);
#pragma unroll
            for (int t = 0; t < 4; ++t) {
                v2f bb = *(const v2f*)(bgrp + (t << 6));
                acc[t] = __builtin_amdgcn_wmma_f32_16x16x4_f32(
                    false, a, false, bb, (short)0, acc[t], false, false);
            }
        }
        if (kc + 1 < NCHUNK) {
            __syncthreads();   // everyone finished reading current chunk
            storeLDS();        // waits for the in-flight global loads here
            __syncthreads();   // next chunk visible to all waves
        }
    }

    // epilogue: C layout -> (P, B, COUT), coalesced across half-wave lanes
    const size_t orow = (size_t)p * BATCH;
#pragma unroll
    for (int t = 0; t < 4; ++t) {
        int cc = n0 + t * 16 + l15;
#pragma unroll
        for (int j = 0; j < 8; ++j) {
            int m = m0 + j + (half << 3);
            out[(orow + m) * COUT + cc] = acc[t][j];
        }
    }
}

// ------------------------------------------------------- BN statistics ----
// Deterministic two-stage reduction over ws_out[(p*64+b)][c].
// 121 blocks x 1024 rows each (121*1024 = 123904 = P*B).
__global__ void stats_partial(const float* __restrict__ in, float* __restrict__ partial) {
    size_t base = (size_t)blockIdx.x * 1024 * COUT;
    float s = 0.f, q = 0.f;
    for (int i = threadIdx.x; i < 1024 * COUT; i += 256) {
        float v = in[base + i];
        s += v;
        q += v * v;
    }
    __shared__ float sh[512];
    sh[threadIdx.x] = s;
    sh[256 + threadIdx.x] = q;
    __syncthreads();
    if (threadIdx.x < 128) {
        float ss = sh[threadIdx.x] + sh[threadIdx.x + 128];
        float qq = sh[256 + threadIdx.x] + sh[384 + threadIdx.x];
        partial[blockIdx.x * 256 + threadIdx.x] = ss;
        partial[blockIdx.x * 256 + 128 + threadIdx.x] = qq;
    }
}

__global__ void stats_final(const float* __restrict__ partial,
                            const float* __restrict__ gamma,
                            const float* __restrict__ beta,
                            float* __restrict__ cA, float* __restrict__ cB) {
    int c = threadIdx.x;  // 128 threads
    float s = 0.f, q = 0.f;
    for (int bk = 0; bk < 121; ++bk) {
        s += partial[bk * 256 + c];
        q += partial[bk * 256 + 128 + c];
    }
    const float n = (float)(P * BATCH);
    float mu = s / n;
    float var = q / n - mu * mu;
    float rs = rsqrtf(var + 1e-5f);
    float a = gamma[c] * rs;
    cA[c] = a;
    cB[c] = beta[c] - mu * a;
}

// ------------------------------------------- transpose + BN + ReLU out ----
// ws_out is (P, BC=8192); d_out is (BC, P). 32x32 LDS tiles, both sides coalesced.
__global__ void transpose_bn_relu(const float* __restrict__ in,
                                  const float* __restrict__ cA,
                                  const float* __restrict__ cB,
                                  float* __restrict__ out) {
    __shared__ float tile[32][33];
    const int p0 = blockIdx.x * 32;
    const int q0 = blockIdx.y * 32;
    const int tx = threadIdx.x;  // 32
    const int ty = threadIdx.y;  // 8
#pragma unroll
    for (int k = 0; k < 4; ++k) {
        int pp = p0 + ty + k * 8;
        if (pp < P) tile[ty + k * 8][tx] = in[(size_t)pp * (BATCH * COUT) + q0 + tx];
    }
    __syncthreads();
#pragma unroll
    for (int k = 0; k < 4; ++k) {
        int q = q0 + ty + k * 8;
        int pp = p0 + tx;
        if (pp < P) {
            int c = q & (COUT - 1);
            float v = tile[tx][ty + k * 8];
            v = v * cA[c] + cB[c];
            out[(size_t)q * P + pp] = fmaxf(v, 0.0f);
        }
    }
}

// ---------------------------------------------------------------- launch ---
extern "C" void kernel_launch(void* const* d_in, const int* in_sizes, int n_in,
                              void* d_out, int out_size, void* d_ws, size_t ws_size,
                              hipStream_t stream) {
    const float* x = (const float*)d_in[0];      // (64,16,48,48)
    const float* kern = (const float*)d_in[1];   // (P,F,COUT)
    const float* gamma = (const float*)d_in[2];  // (128,)
    const float* beta = (const float*)d_in[3];   // (128,)
    float* out = (float*)d_out;                  // (64,128,44,44)

    float* wsf = (float*)d_ws;
    float* scale = wsf;                      // 64
    float* cA = wsf + 64;                    // 128
    float* cB = wsf + 192;                   // 128
    float* partial = wsf + 320;              // 121*256 = 30976
    float* gout = wsf + 320 + 30976;         // P*BATCH*COUT floats (~63.4 MB)

    norm_kernel<<<BATCH, 256, 0, stream>>>(x, scale);
    gemm_kernel<<<P, 256, 0, stream>>>(x, kern, scale, gout);
    stats_partial<<<121, 256, 0, stream>>>(gout, partial);
    stats_final<<<1, 128, 0, stream>>>(partial, gamma, beta, cA, cB);
    dim3 tgrid((P + 31) / 32, (BATCH * COUT) / 32);
    dim3 tblk(32, 8);
    transpose_bn_relu<<<tgrid, tblk, 0, stream>>>(gout, cA, cB, out);
}